// SupervisedContrastiveMultimodalAlignmentLoss_68856915690212
// MI455X (gfx1250) — compile-verified
//
#include <hip/hip_runtime.h>
#include <math.h>

// Shapes from the reference
#define A_N 256
#define T_N 32
#define D_N 512
#define B_N 256
#define V_N 64
#define NEG_INF_F (-1e30f)
#define LDS_STRIDE 129   // 128 cols + 1 pad -> conflict-free column walks

typedef __attribute__((ext_vector_type(2))) float v2f;
typedef __attribute__((ext_vector_type(8))) float v8f;

// ---------------------------------------------------------------------------
// Masked softmax weights over tokens (wA) and frames (wB)
// blocks 0..255 -> wA rows (32 elems), blocks 256..511 -> wB rows (64 elems)
// ---------------------------------------------------------------------------
__global__ __launch_bounds__(64) void weights_kernel(
    const int* __restrict__ tokens, const float* __restrict__ tfw,
    const float* __restrict__ vfw, const float* __restrict__ vmask,
    float* __restrict__ wA, float* __restrict__ wB) {
  __shared__ float red[64];
  const int tid = threadIdx.x;
  const int bid = blockIdx.x;
  float x = NEG_INF_F;
  if (bid < A_N) {
    if (tid < T_N) {
      const int idx = bid * T_N + tid;
      x = (tokens[idx] != 0) ? tfw[idx] : NEG_INF_F;
    }
  } else {
    const int idx = (bid - A_N) * V_N + tid;
    x = (vmask[idx] > 0.f) ? vfw[idx] : NEG_INF_F;
  }
  red[tid] = x; __syncthreads();
  for (int s = 32; s > 0; s >>= 1) { if (tid < s) red[tid] = fmaxf(red[tid], red[tid + s]); __syncthreads(); }
  const float mx = red[0]; __syncthreads();
  float e = __expf(x - mx);
  if (bid < A_N && tid >= T_N) e = 0.f;
  red[tid] = e; __syncthreads();
  for (int s = 32; s > 0; s >>= 1) { if (tid < s) red[tid] += red[tid + s]; __syncthreads(); }
  const float inv = 1.0f / red[0];
  if (bid < A_N) { if (tid < T_N) wA[bid * T_N + tid] = e * inv; }
  else            wB[(bid - A_N) * V_N + tid] = e * inv;
}

// ---------------------------------------------------------------------------
// cnt[i] = #labels equal to labels[i]  (rowsum of eq-matrix)
// ---------------------------------------------------------------------------
__global__ void cnt_kernel(const int* __restrict__ labels, float* __restrict__ cnt) {
  const int i = threadIdx.x;
  const int li = labels[i];
  int c = 0;
  for (int j = 0; j < A_N; ++j) c += (labels[j] == li) ? 1 : 0;
  cnt[i] = (float)c;
}

// ---------------------------------------------------------------------------
// score[a,b] via fp32 WMMA (exact): one workgroup per (a, b-pair) computes a
// 32x128 logits tile with K=512, then reduces max_v / max_t with masks.
// Fragment layout of V_WMMA_F32_16X16X4_F32: lane holds 2 consecutive floats
// of a row for both A (M-major) and B (video is N-major over K) -> no
// transpose, direct 8B global loads, reuse served from the 192MB L2.
// ---------------------------------------------------------------------------
__global__ __launch_bounds__(128) void score_kernel(
    const float* __restrict__ text, const float* __restrict__ video,
    const int* __restrict__ tokens, const float* __restrict__ vmask,
    const float* __restrict__ wA, const float* __restrict__ wB,
    float* __restrict__ score) {
  __shared__ float tile[T_N * LDS_STRIDE];
  __shared__ float red[128];
  __shared__ float maskA_sh[T_N];
  __shared__ float a2b_sh[2];

  const int a    = blockIdx.x;       // text item: rows a*32 .. a*32+31
  const int bblk = blockIdx.y;       // pair of videos: cols bblk*128 .. +127
  const int tid  = threadIdx.x;
  const int wv   = tid >> 5;         // wave 0..3 -> 32-col slice
  const int lane = tid & 31;
  const int lr   = lane & 15;        // fragment row (M for A, N for B)
  const int lk   = (lane >> 4) << 1; // fragment K sub-offset {0,2}

  const float* pa0 = text  + (size_t)(a * T_N + lr) * D_N + lk;
  const float* pa1 = pa0 + 16 * D_N;
  const float* pb0 = video + (size_t)(bblk * 128 + wv * 32 + lr) * D_N + lk;
  const float* pb1 = pb0 + 16 * D_N;

  v8f c00 = {}, c01 = {}, c10 = {}, c11 = {};
#pragma unroll 4
  for (int k = 0; k < D_N; k += 4) {
    const v2f af0 = *(const v2f*)(pa0 + k);
    const v2f af1 = *(const v2f*)(pa1 + k);
    const v2f bf0 = *(const v2f*)(pb0 + k);
    const v2f bf1 = *(const v2f*)(pb1 + k);
    c00 = __builtin_amdgcn_wmma_f32_16x16x4_f32(false, af0, false, bf0, (short)0, c00, false, false);
    c01 = __builtin_amdgcn_wmma_f32_16x16x4_f32(false, af0, false, bf1, (short)0, c01, false, false);
    c10 = __builtin_amdgcn_wmma_f32_16x16x4_f32(false, af1, false, bf0, (short)0, c10, false, false);
    c11 = __builtin_amdgcn_wmma_f32_16x16x4_f32(false, af1, false, bf1, (short)0, c11, false, false);
  }

  // C layout: VGPR r -> (M = r + 8*(lane>=16), N = lane&15)
  const int rbase = (lane >> 4) << 3;
  const int cbase = wv * 32 + lr;
#pragma unroll
  for (int r = 0; r < 8; ++r) {
    tile[(rbase + r)      * LDS_STRIDE + cbase]      = c00[r];
    tile[(rbase + r)      * LDS_STRIDE + cbase + 16] = c01[r];
    tile[(16 + rbase + r) * LDS_STRIDE + cbase]      = c10[r];
    tile[(16 + rbase + r) * LDS_STRIDE + cbase + 16] = c11[r];
  }
  if (tid < T_N) maskA_sh[tid] = (tokens[a * T_N + tid] != 0) ? 1.f : 0.f;
  __syncthreads();

  // a2b: per (t, bb) max over the 64 frames of video b, weighted by wA
  float av = 0.f;
  if (tid < 64) {
    const int t = tid & 31, bb = tid >> 5;
    const float mA = maskA_sh[t];
    float mxv = NEG_INF_F;
    for (int c = 0; c < V_N; ++c) {
      const float mB = vmask[bblk * 128 + bb * V_N + c];
      mxv = fmaxf(mxv, tile[t * LDS_STRIDE + bb * V_N + c] * mA * mB);
    }
    av = wA[a * T_N + t] * mxv;
  }
  red[tid] = av;
  __syncthreads();
  if (tid < 2) {
    float s = 0.f;
    for (int i = 0; i < 32; ++i) s += red[tid * 32 + i];
    a2b_sh[tid] = s;
  }
  __syncthreads();

  // b2a: per column max over the 32 tokens, weighted by wB
  {
    const float mB = vmask[bblk * 128 + tid];
    float mxv = NEG_INF_F;
    for (int t = 0; t < T_N; ++t)
      mxv = fmaxf(mxv, tile[t * LDS_STRIDE + tid] * maskA_sh[t] * mB);
    red[tid] = wB[bblk * 128 + tid] * mxv;
  }
  __syncthreads();
  if (tid < 2) {
    float s = 0.f;
    for (int i = 0; i < 64; ++i) s += red[tid * 64 + i];
    score[a * B_N + bblk * 2 + tid] = 0.5f * (a2b_sh[tid] + s);
  }
}

// ---------------------------------------------------------------------------
// loss1[a] (blocks 0..255, log_softmax over b) and loss2[b] (blocks 256..511,
// log_softmax over a): loss = lse*sum(p) - sum(p*s), p = eq / cnt[row-of-a]
// ---------------------------------------------------------------------------
__global__ __launch_bounds__(256) void loss_kernel(
    const float* __restrict__ score, const int* __restrict__ labels,
    const float* __restrict__ cnt, float* __restrict__ lossv) {
  __shared__ float red[256];
  const int tid = threadIdx.x;
  const int bid = blockIdx.x;
  const bool rowmode = bid < A_N;
  const int idx = rowmode ? bid : bid - A_N;
  const float s = (rowmode ? score[idx * B_N + tid] : score[tid * B_N + idx]) * (1.0f / 0.07f);
  const float eqv = (labels[tid] == labels[idx]) ? 1.f : 0.f;

  red[tid] = s; __syncthreads();
  for (int st = 128; st > 0; st >>= 1) { if (tid < st) red[tid] = fmaxf(red[tid], red[tid + st]); __syncthreads(); }
  const float mx = red[0]; __syncthreads();
  red[tid] = __expf(s - mx); __syncthreads();
  for (int st = 128; st > 0; st >>= 1) { if (tid < st) red[tid] += red[tid + st]; __syncthreads(); }
  const float lse = __logf(red[0]) + mx; __syncthreads();

  // p = eq / clip(rowsum(eq),1); row index of eq is always the "a" axis
  const float p = eqv / fmaxf(rowmode ? cnt[idx] : cnt[tid], 1.0f);
  red[tid] = p; __syncthreads();
  for (int st = 128; st > 0; st >>= 1) { if (tid < st) red[tid] += red[tid + st]; __syncthreads(); }
  const float sum_p = red[0]; __syncthreads();
  red[tid] = p * s; __syncthreads();
  for (int st = 128; st > 0; st >>= 1) { if (tid < st) red[tid] += red[tid + st]; __syncthreads(); }
  if (tid == 0) lossv[bid] = lse * sum_p - red[0];
}

// ---------------------------------------------------------------------------
// mean over concat(loss1, loss2)  (512 values) -> d_out[0]
// ---------------------------------------------------------------------------
__global__ __launch_bounds__(512) void final_kernel(
    const float* __restrict__ lossv, float* __restrict__ out) {
  __shared__ float red[512];
  const int tid = threadIdx.x;
  red[tid] = lossv[tid];
  __syncthreads();
  for (int st = 256; st > 0; st >>= 1) { if (tid < st) red[tid] += red[tid + st]; __syncthreads(); }
  if (tid == 0) out[0] = red[0] * (1.0f / 512.0f);
}

extern "C" void kernel_launch(void* const* d_in, const int* in_sizes, int n_in,
                              void* d_out, int out_size, void* d_ws, size_t ws_size,
                              hipStream_t stream) {
  (void)in_sizes; (void)n_in; (void)out_size; (void)ws_size;
  const int*   tokens = (const int*)  d_in[0];  // [256,32] int32
  const float* text   = (const float*)d_in[1];  // [256,32,512]
  const float* tfw    = (const float*)d_in[2];  // [256,32,1]
  const float* video  = (const float*)d_in[3];  // [256,64,512]
  const float* vfw    = (const float*)d_in[4];  // [256,64,1]
  const float* vmask  = (const float*)d_in[5];  // [256,64]
  const int*   labels = (const int*)  d_in[6];  // [256]
  float* out = (float*)d_out;

  float* ws    = (float*)d_ws;
  float* wA    = ws;                     // 256*32   = 8192
  float* wB    = ws + 8192;              // 256*64   = 16384
  float* scoreb = ws + 24576;            // 256*256  = 65536
  float* cnt   = ws + 90112;             // 256
  float* lossv = ws + 90368;             // 512
  // total 90880 floats (~356 KB) of workspace

  weights_kernel<<<512, 64, 0, stream>>>(tokens, tfw, vfw, vmask, wA, wB);
  cnt_kernel<<<1, 256, 0, stream>>>(labels, cnt);
  score_kernel<<<dim3(A_N, B_N / 2), 128, 0, stream>>>(text, video, tokens, vmask, wA, wB, scoreb);
  loss_kernel<<<512, 256, 0, stream>>>(scoreb, labels, cnt, lossv);
  final_kernel<<<1, 512, 0, stream>>>(lossv, out);
}